// SparseMMLayer_53257594470703
// MI455X (gfx1250) — compile-verified
//
#include <hip/hip_runtime.h>

typedef __attribute__((ext_vector_type(2))) float v2f;
typedef __attribute__((ext_vector_type(8))) float v8f;

#define BB   32
#define MDIM 4096
#define KDIM 64
#define NDIM 4096
#define TOPK 32

// ---------------------------------------------------------------------------
// Kernel 1: transpose B [BB][K][N] -> Bt [BB][N][K] (tiled via LDS, 32x33 pad)
// Turns the hot kernel's 16KB-stride 4B gathers into contiguous 8B loads.
// ---------------------------------------------------------------------------
__global__ __launch_bounds__(256) void bt_transpose(const float* __restrict__ B,
                                                    float* __restrict__ Bt) {
    __shared__ float tile[32][33];
    const int tx = threadIdx.x;          // 0..31
    const int ty = threadIdx.y;          // 0..7
    const int n0 = blockIdx.x * 32;
    const int k0 = blockIdx.y * 32;
    const int b  = blockIdx.z;

    const float* src = B  + (size_t)b * KDIM * NDIM;
    float*       dst = Bt + (size_t)b * NDIM * KDIM;

#pragma unroll
    for (int i = 0; i < 4; ++i) {
        const int k = k0 + ty + i * 8;
        tile[ty + i * 8][tx] = src[(size_t)k * NDIM + (n0 + tx)];
    }
    __syncthreads();
#pragma unroll
    for (int i = 0; i < 4; ++i) {
        const int n = n0 + ty + i * 8;
        dst[(size_t)n * KDIM + (k0 + tx)] = tile[tx][ty + i * 8];
    }
}

// ---------------------------------------------------------------------------
// Kernel 2: one wave32 per (b, m).
//   D = A_op(16x4, broadcast A-row chunk) x B_op(4x16, gathered Bt columns)
//   => every row of D equals c[0..15]; lane L reads c[L%16] from acc[0].
// acc0 covers t = 0..15, acc1 covers t = 16..31 (topk = 32 = 2 tiles).
// ---------------------------------------------------------------------------
template <bool TRANSPOSED>
__global__ __launch_bounds__(256) void sparse_mm_wmma(const float* __restrict__ A,
                                                      const float* __restrict__ Bsrc,
                                                      const int* __restrict__ index,
                                                      float* __restrict__ C) {
    const int lane = threadIdx.x & 31;
    const int wave = threadIdx.x >> 5;
    const int gid  = blockIdx.x * 8 + wave;    // flat (b*MDIM + m), exact grid
    const int b    = gid >> 12;                // MDIM == 4096 == 2^12
    const int hi   = lane >> 4;                // half-wave: selects K pair (2hi, 2hi+1)
    const int tl   = lane & 15;                // column (t within tile)

    // Gathered column indices for the two t-tiles this lane's column serves.
    const int i0 = index[(size_t)gid * TOPK + tl];
    const int i1 = index[(size_t)gid * TOPK + 16 + tl];

    const float* Arow = A + (size_t)gid * KDIM;
    const float* r0;
    const float* r1;
    if (TRANSPOSED) {
        r0 = Bsrc + ((size_t)b * NDIM + i0) * KDIM;   // contiguous K row
        r1 = Bsrc + ((size_t)b * NDIM + i1) * KDIM;
    } else {
        r0 = Bsrc + (size_t)b * KDIM * NDIM + i0;     // stride-N column
        r1 = Bsrc + (size_t)b * KDIM * NDIM + i1;
    }

    v8f acc0 = {};
    v8f acc1 = {};

#pragma unroll
    for (int c = 0; c < KDIM / 4; ++c) {
        const int kb = 4 * c + 2 * hi;      // this half-wave's K pair in chunk c

        // A-operand: broadcast A-row pair (same for all 16 M rows).
        v2f av = *(const v2f*)(Arow + kb);

        // B-operand: gathered Bt rows (lane tl -> output column tl).
        v2f b0, b1;
        if (TRANSPOSED) {
            b0 = *(const v2f*)(r0 + kb);
            b1 = *(const v2f*)(r1 + kb);
        } else {
            b0.x = r0[(size_t)kb * NDIM];
            b0.y = r0[(size_t)(kb + 1) * NDIM];
            b1.x = r1[(size_t)kb * NDIM];
            b1.y = r1[(size_t)(kb + 1) * NDIM];
        }

        // 8 args: (neg_a, A, neg_b, B, c_mod, C, reuse_a, reuse_b)
        acc0 = __builtin_amdgcn_wmma_f32_16x16x4_f32(false, av, false, b0,
                                                     (short)0, acc0, false, false);
        acc1 = __builtin_amdgcn_wmma_f32_16x16x4_f32(false, av, false, b1,
                                                     (short)0, acc1, false, false);
    }

    // All D rows identical => acc[0] at lane L is c[L%16] of its tile.
    const float out = hi ? acc1[0] : acc0[0];
    C[(size_t)gid * TOPK + lane] = out;      // coalesced 128B store per wave
}

// ---------------------------------------------------------------------------
extern "C" void kernel_launch(void* const* d_in, const int* in_sizes, int n_in,
                              void* d_out, int out_size, void* d_ws, size_t ws_size,
                              hipStream_t stream) {
    const float* A   = (const float*)d_in[0];
    const float* B   = (const float*)d_in[1];
    const int*   idx = (const int*)d_in[2];
    float*       C   = (float*)d_out;

    const size_t btBytes = (size_t)BB * KDIM * NDIM * sizeof(float);

    const dim3 mblk(256);
    const dim3 mgrid((BB * MDIM) / 8);       // 8 waves per block, exact cover

    if (d_ws != nullptr && ws_size >= btBytes) {
        float* Bt = (float*)d_ws;
        const dim3 tb(32, 8);
        const dim3 tg(NDIM / 32, KDIM / 32, BB);
        hipLaunchKernelGGL(bt_transpose, tg, tb, 0, stream, B, Bt);
        hipLaunchKernelGGL((sparse_mm_wmma<true>), mgrid, mblk, 0, stream, A, Bt, idx, C);
    } else {
        hipLaunchKernelGGL((sparse_mm_wmma<false>), mgrid, mblk, 0, stream, A, B, idx, C);
    }
}